// ProposalLayer_71502615543912
// MI455X (gfx1250) — compile-verified
//
#include <hip/hip_runtime.h>
#include <hip/hip_bf16.h>
#include <math.h>

// ---- problem constants (from reference) ----
#define IMG_H   100
#define IMG_W   152
#define NA      9
#define NTOT    (IMG_H * IMG_W * NA)      // 136800
#define NSORT   262144                    // next pow2 >= NTOT
#define PRE_NMS 6000
#define POST_NMS 300
#define CB      ((PRE_NMS + 31) / 32)     // 188 column words of 32 bits
#define NMS_TH  0.7f
#define FEAT    16.0f

// bitonic tiling: 4096 keys (32KB LDS) per block, 512 threads
#define TILE     4096
#define TTHREADS 512

// base anchors: all have center (8,8); widths/heights per anchor index
__constant__ float c_aw[NA] = {184.f, 368.f, 736.f, 128.f, 256.f, 512.f,  88.f, 176.f, 352.f};
__constant__ float c_ah[NA] = { 96.f, 192.f, 384.f, 128.f, 256.f, 512.f, 176.f, 352.f, 704.f};

// float -> order-preserving unsigned key (NaN-free inputs here)
__device__ __forceinline__ unsigned sortable_bits(float f) {
    unsigned u = __float_as_uint(f);
    return (u & 0x80000000u) ? ~u : (u | 0x80000000u);
}

// ------------------------------------------------------------------
// 1) decode boxes + build sort keys
// ------------------------------------------------------------------
__global__ void decode_kernel(const float* __restrict__ probs,
                              const float* __restrict__ deltas,
                              const float* __restrict__ img_info,
                              float4* __restrict__ boxes,
                              unsigned long long* __restrict__ keys)
{
#pragma clang fp contract(off)
    int t = blockIdx.x * blockDim.x + threadIdx.x;
    if (t >= NSORT) return;
    if (t >= NTOT) { keys[t] = 0ull; return; }

    const int HW = IMG_H * IMG_W;
    int a  = t % NA;
    int hw = t / NA;
    int w  = hw % IMG_W;
    int h  = hw / IMG_W;
    int pix = h * IMG_W + w;

    float score = probs[(NA + a) * HW + pix];
    float dx = deltas[(a * 4 + 0) * HW + pix];
    float dy = deltas[(a * 4 + 1) * HW + pix];
    float dw = deltas[(a * 4 + 2) * HW + pix];
    float dh = deltas[(a * 4 + 3) * HW + pix];
    dw = fminf(fmaxf(dw, -10.0f), 10.0f);
    dh = fminf(fmaxf(dh, -10.0f), 10.0f);

    float aw = c_aw[a], ah = c_ah[a];
    float acx = 8.0f + (float)w * FEAT;
    float acy = 8.0f + (float)h * FEAT;

    float pcx = dx * aw + acx;
    float pcy = dy * ah + acy;
    float pw  = expf(dw) * aw;
    float ph  = expf(dh) * ah;

    float x1 = pcx - 0.5f * pw;
    float y1 = pcy - 0.5f * ph;
    float x2 = pcx + 0.5f * pw;
    float y2 = pcy + 0.5f * ph;

    float im_h = img_info[0], im_w = img_info[1], sc = img_info[2];
    x1 = fminf(fmaxf(x1, 0.0f), im_w - 1.0f);
    y1 = fminf(fmaxf(y1, 0.0f), im_h - 1.0f);
    x2 = fminf(fmaxf(x2, 0.0f), im_w - 1.0f);
    y2 = fminf(fmaxf(y2, 0.0f), im_h - 1.0f);

    float ws = x2 - x1 + 1.0f;
    float hs = y2 - y1 + 1.0f;
    float min_size = FEAT * sc;
    bool valid = (ws >= min_size) && (hs >= min_size);

    boxes[t] = make_float4(x1, y1, x2, y2);

    float masked = valid ? score : -INFINITY;
    unsigned hi = sortable_bits(masked);
    keys[t] = ((unsigned long long)hi << 32) | (unsigned)(~(unsigned)t);
}

// ------------------------------------------------------------------
// 2) bitonic sort (descending) of NSORT 64-bit keys
//    comparator: when ((gi & k) == 0), larger key goes first
// ------------------------------------------------------------------

// in-LDS sort of a 4096 tile: all stages k = 2..TILE
__global__ __launch_bounds__(TTHREADS)
void bitonic_tile_sort(unsigned long long* __restrict__ keys)
{
    __shared__ unsigned long long s[TILE];
    int base = blockIdx.x * TILE;
    for (int e = threadIdx.x; e < TILE; e += TTHREADS) s[e] = keys[base + e];
    __syncthreads();
    for (int k = 2; k <= TILE; k <<= 1) {
        for (int j = k >> 1; j >= 1; j >>= 1) {
            for (int p = threadIdx.x; p < TILE / 2; p += TTHREADS) {
                int i = ((p & ~(j - 1)) << 1) | (p & (j - 1));
                int l = i | j;
                unsigned long long a = s[i], b = s[l];
                if ((((base + i) & k) == 0) == (a < b)) { s[i] = b; s[l] = a; }
            }
            __syncthreads();
        }
    }
    for (int e = threadIdx.x; e < TILE; e += TTHREADS) keys[base + e] = s[e];
}

// one global compare-exchange step (j >= TILE); NSORT/2 pairs
__global__ void bitonic_global_step(unsigned long long* __restrict__ keys, int k, int j)
{
    int p = blockIdx.x * blockDim.x + threadIdx.x;
    int i = ((p & ~(j - 1)) << 1) | (p & (j - 1));
    int l = i | j;
    unsigned long long a = keys[i], b = keys[l];
    if (((i & k) == 0) == (a < b)) { keys[i] = b; keys[l] = a; }
}

// fused in-LDS steps j = TILE/2 .. 1 for a given k ( > TILE )
__global__ __launch_bounds__(TTHREADS)
void bitonic_fused(unsigned long long* __restrict__ keys, int k)
{
    __shared__ unsigned long long s[TILE];
    int base = blockIdx.x * TILE;
    for (int e = threadIdx.x; e < TILE; e += TTHREADS) s[e] = keys[base + e];
    __syncthreads();
    for (int j = TILE >> 1; j >= 1; j >>= 1) {
        for (int p = threadIdx.x; p < TILE / 2; p += TTHREADS) {
            int i = ((p & ~(j - 1)) << 1) | (p & (j - 1));
            int l = i | j;
            unsigned long long a = s[i], b = s[l];
            if ((((base + i) & k) == 0) == (a < b)) { s[i] = b; s[l] = a; }
        }
        __syncthreads();
    }
    for (int e = threadIdx.x; e < TILE; e += TTHREADS) keys[base + e] = s[e];
}

// ------------------------------------------------------------------
// 3) gather top PRE_NMS boxes + alive flags
// ------------------------------------------------------------------
__global__ void gather_kernel(const unsigned long long* __restrict__ keys,
                              const float4* __restrict__ boxes,
                              float4* __restrict__ topBoxes,
                              int* __restrict__ alive)
{
    int t = blockIdx.x * blockDim.x + threadIdx.x;
    if (t >= PRE_NMS) return;
    unsigned long long key = keys[t];
    unsigned hi = (unsigned)(key >> 32);
    unsigned idx = ~(unsigned)key;
    int ok = ((hi & 0x80000000u) != 0u) && (idx < NTOT);
    float4 b = make_float4(0.f, 0.f, 0.f, 0.f);
    if (ok) b = boxes[idx];
    topBoxes[t] = b;
    alive[t] = ok;
}

// ------------------------------------------------------------------
// 4) NMS suppression bitmask: one wave32 per (rowblock, colblock)
//    column boxes staged into LDS with gfx1250 async global->LDS copy
// ------------------------------------------------------------------
__global__ __launch_bounds__(32)
void nms_mask_kernel(const float4* __restrict__ topBoxes,
                     unsigned* __restrict__ mask)
{
#pragma clang fp contract(off)
    int cb = blockIdx.x;   // column block
    int rb = blockIdx.y;   // row block
    int t  = threadIdx.x;  // 0..31

    if (cb < rb) {  // strictly lower triangle: j < i always -> zero word
        int i = rb * 32 + t;
        if (i < PRE_NMS) mask[(size_t)i * CB + cb] = 0u;
        return;
    }

    __shared__ float4 cbox[32];

    int col  = cb * 32 + t;
    int colc = col < PRE_NMS ? col : (PRE_NMS - 1);

    // gfx1250 async path: per-lane 16B global -> LDS copy, tracked by
    // ASYNCcnt, then wait. All 32 lanes of the single wave are active here.
    {
        unsigned lds_off = (unsigned)(unsigned long long)(&cbox[t]);
        unsigned long long gaddr = (unsigned long long)(&topBoxes[colc]);
        asm volatile("global_load_async_to_lds_b128 %0, %1, off"
                     :: "v"(lds_off), "v"(gaddr) : "memory");
        asm volatile("s_wait_asynccnt 0x0" ::: "memory");
    }
    __builtin_amdgcn_wave_barrier();

    int i = rb * 32 + t;
    if (i >= PRE_NMS) return;

    float4 r = topBoxes[i];
    float rarea = (r.z - r.x + 1.0f) * (r.w - r.y + 1.0f);

    unsigned bits = 0u;
    #pragma unroll 4
    for (int jj = 0; jj < 32; ++jj) {
        int j = cb * 32 + jj;
        if (j > i && j < PRE_NMS) {
            float4 c = cbox[jj];
            float carea = (c.z - c.x + 1.0f) * (c.w - c.y + 1.0f);
            float ix1 = fmaxf(r.x, c.x);
            float iy1 = fmaxf(r.y, c.y);
            float ix2 = fminf(r.z, c.z);
            float iy2 = fminf(r.w, c.w);
            float iw = fmaxf(ix2 - ix1 + 1.0f, 0.0f);
            float ih = fmaxf(iy2 - iy1 + 1.0f, 0.0f);
            float inter = iw * ih;
            float iou = inter / (rarea + carea - inter);
            if (iou > NMS_TH) bits |= (1u << jj);
        }
    }
    mask[(size_t)i * CB + cb] = bits;
}

// ------------------------------------------------------------------
// 5) greedy NMS reduce (single wave32) + write (300,5) output
// ------------------------------------------------------------------
__global__ __launch_bounds__(32)
void nms_reduce_kernel(const unsigned* __restrict__ mask,
                       const int* __restrict__ alive,
                       const float4* __restrict__ topBoxes,
                       float* __restrict__ out)
{
    int lane = threadIdx.x;           // 32 lanes, lane owns words w with (w&31)==lane
    unsigned removed0 = 0, removed1 = 0, removed2 = 0,
             removed3 = 0, removed4 = 0, removed5 = 0;
    __shared__ int keepList[POST_NMS];
    int count = 0;

    for (int i = 0; i < PRE_NMS; ++i) {
        int word  = i >> 5;
        int bit   = i & 31;
        int owner = word & 31;        // wave-uniform
        int q     = word >> 5;        // wave-uniform, 0..5
        unsigned myv;
        switch (q) {
            case 0: myv = removed0; break;
            case 1: myv = removed1; break;
            case 2: myv = removed2; break;
            case 3: myv = removed3; break;
            case 4: myv = removed4; break;
            default: myv = removed5; break;
        }
        unsigned v = (unsigned)__shfl((int)myv, owner, 32);
        bool kept = (alive[i] != 0) && (((v >> bit) & 1u) == 0u);
        if (kept) {
            if (lane == 0) keepList[count] = i;
            count++;
            if (count >= POST_NMS) break;
            const unsigned* row = mask + (size_t)i * CB;
            int w0 = lane;
            removed0 |= row[w0];
            removed1 |= row[w0 + 32];
            removed2 |= row[w0 + 64];
            removed3 |= row[w0 + 96];
            removed4 |= row[w0 + 128];
            if (w0 + 160 < CB) removed5 |= row[w0 + 160];
        }
    }
    __syncthreads();

    for (int k = lane; k < POST_NMS; k += 32) {
        float4 b = make_float4(0.f, 0.f, 0.f, 0.f);
        if (k < count) b = topBoxes[keepList[k]];
        out[k * 5 + 0] = 0.0f;
        out[k * 5 + 1] = b.x;
        out[k * 5 + 2] = b.y;
        out[k * 5 + 3] = b.z;
        out[k * 5 + 4] = b.w;
    }
}

// ------------------------------------------------------------------
// launcher
// ------------------------------------------------------------------
extern "C" void kernel_launch(void* const* d_in, const int* in_sizes, int n_in,
                              void* d_out, int out_size, void* d_ws, size_t ws_size,
                              hipStream_t stream)
{
    const float* probs    = (const float*)d_in[0];
    const float* deltas   = (const float*)d_in[1];
    const float* img_info = (const float*)d_in[2];
    float* out = (float*)d_out;

    char* ws = (char*)d_ws;
    size_t off = 0;
    unsigned long long* keys = (unsigned long long*)(ws + off); off += (size_t)NSORT * 8;
    float4* boxes            = (float4*)(ws + off);             off += (size_t)NTOT * 16;
    float4* topBoxes         = (float4*)(ws + off);             off += (size_t)PRE_NMS * 16;
    int* alive               = (int*)(ws + off);                off += (size_t)PRE_NMS * 4;
    unsigned* mask           = (unsigned*)(ws + off);           off += (size_t)PRE_NMS * CB * 4;

    // 1) decode boxes + keys (padded to NSORT)
    decode_kernel<<<NSORT / 256, 256, 0, stream>>>(probs, deltas, img_info, boxes, keys);

    // 2) bitonic sort, descending
    bitonic_tile_sort<<<NSORT / TILE, TTHREADS, 0, stream>>>(keys);   // k = 2..TILE
    for (int k = TILE * 2; k <= NSORT; k <<= 1) {
        for (int j = k >> 1; j >= TILE; j >>= 1)
            bitonic_global_step<<<(NSORT / 2) / 256, 256, 0, stream>>>(keys, k, j);
        bitonic_fused<<<NSORT / TILE, TTHREADS, 0, stream>>>(keys, k); // j = TILE/2..1
    }

    // 3) gather top PRE_NMS
    gather_kernel<<<(PRE_NMS + 255) / 256, 256, 0, stream>>>(keys, boxes, topBoxes, alive);

    // 4) suppression bitmask
    dim3 mgrid(CB, CB);
    nms_mask_kernel<<<mgrid, 32, 0, stream>>>(topBoxes, mask);

    // 5) greedy reduce + output
    nms_reduce_kernel<<<1, 32, 0, stream>>>(mask, alive, topBoxes, out);
}